// reviewContext_15642270892408
// MI455X (gfx1250) — compile-verified
//
#include <hip/hip_runtime.h>
#include <hip/hip_bf16.h>

#define D_MODEL 256
#define NCODES  32
#define L_SEQ   2048
#define BATCH   64
#define LDSTR   260     // padded LDS row stride in floats (260 % 64 = 4 -> conflict-free)
#define ROWS_A  64      // rows per workgroup in scores kernel (4 waves x 16 rows)
#define ROWS_C  64      // rows per double-buffered chunk in context kernel
#define NCHUNK  8       // 512-row segment / 64-row chunks

typedef __attribute__((ext_vector_type(2))) float v2f;
typedef __attribute__((ext_vector_type(8))) float v8f;

__device__ __forceinline__ v8f wmma_f32(v2f a, v2f b, v8f c) {
    return __builtin_amdgcn_wmma_f32_16x16x4_f32(
        /*neg_a=*/false, a, /*neg_b=*/false, b,
        /*c_mod=*/(short)0, c, /*reuse_a=*/false, /*reuse_b=*/false);
}

// Async global->LDS copy of 16 bytes per lane (GLOBAL_LOAD_ASYNC_TO_LDS_B128, GV mode).
// LDS destination = low 32 bits of the generic shared-memory pointer (LDS byte offset).
__device__ __forceinline__ void async_copy_b128(const float* gsrc, float* ldst) {
    unsigned lds = (unsigned)(uintptr_t)ldst;
    unsigned long long ga = (unsigned long long)(uintptr_t)gsrc;
    asm volatile("global_load_async_to_lds_b128 %0, %1, off"
                 :: "v"(lds), "v"(ga) : "memory");
}

__device__ __forceinline__ void wait_asynccnt0() {
#if __has_builtin(__builtin_amdgcn_s_wait_asynccnt)
    __builtin_amdgcn_s_wait_asynccnt(0);
#else
    asm volatile("s_wait_asynccnt 0x0" ::: "memory");
#endif
}

// ---------------- Kernel 1: S[b,l,k] = X[b,l,:] . W[k,:]  (WMMA f32, async staging) ----------------
__global__ __launch_bounds__(128) void rc_scores_kernel(const float* __restrict__ X,
                                                        const float* __restrict__ W,
                                                        float* __restrict__ S) {
    extern __shared__ float smem[];
    float* Xs = smem;                       // 64 x LDSTR
    float* Ws = smem + ROWS_A * LDSTR;      // 32 x LDSTR

    const int b    = blockIdx.x;
    const int row0 = blockIdx.y * ROWS_A;
    const int tid  = threadIdx.x;

    const float* Xg = X + ((size_t)b * L_SEQ + row0) * D_MODEL;
    for (int idx = tid; idx < ROWS_A * (D_MODEL / 4); idx += 128) {
        int r = idx >> 6, c4 = idx & 63;
        async_copy_b128(Xg + (size_t)r * D_MODEL + c4 * 4, Xs + r * LDSTR + c4 * 4);
    }
    for (int idx = tid; idx < NCODES * (D_MODEL / 4); idx += 128) {
        int r = idx >> 6, c4 = idx & 63;
        async_copy_b128(W + (size_t)r * D_MODEL + c4 * 4, Ws + r * LDSTR + c4 * 4);
    }
    wait_asynccnt0();
    __syncthreads();

    const int wave = tid >> 5;    // 0..3
    const int lane = tid & 31;
    const int half = lane >> 4;   // 0: K pair {0,1}, 1: K pair {2,3}
    const int lm   = lane & 15;

    const float* Arow  = Xs + (wave * 16 + lm) * LDSTR + half * 2;  // A: M=lm, K=k0+2h+{0,1}
    const float* Brow0 = Ws + lm * LDSTR + half * 2;                // B: N=lm      (codes 0..15)
    const float* Brow1 = Ws + (16 + lm) * LDSTR + half * 2;         // B: N=16+lm   (codes 16..31)

    v8f acc0 = {}, acc1 = {};
    for (int k0 = 0; k0 < D_MODEL; k0 += 4) {
        v2f a  = *(const v2f*)(Arow + k0);
        v2f b0 = *(const v2f*)(Brow0 + k0);
        v2f b1 = *(const v2f*)(Brow1 + k0);
        acc0 = wmma_f32(a, b0, acc0);
        acc1 = wmma_f32(a, b1, acc1);
    }

    // D layout: VGPR r -> M = r + 8*half, N = lm
    float* Sg = S + ((size_t)b * L_SEQ + row0 + wave * 16) * NCODES;
    for (int r = 0; r < 8; ++r) {
        int m = r + half * 8;
        Sg[(size_t)m * NCODES + lm]      = acc0[r];
        Sg[(size_t)m * NCODES + 16 + lm] = acc1[r];
    }
}

// ---------------- Kernel 2: softmax over L (axis=1) per (b,k), in place ----------------
__global__ __launch_bounds__(256) void rc_softmax_kernel(float* __restrict__ S) {
    __shared__ float red[8][NCODES];
    const int b = blockIdx.x;
    const int k = threadIdx.x & 31;
    const int g = threadIdx.x >> 5;           // 8 row-groups
    float* Sb = S + (size_t)b * L_SEQ * NCODES;

    float mx = -3.402823466e38f;
    for (int l = g; l < L_SEQ; l += 8) mx = fmaxf(mx, Sb[(size_t)l * NCODES + k]);
    red[g][k] = mx;
    __syncthreads();
    if (g == 0) {
        float m = red[0][k];
        for (int i = 1; i < 8; ++i) m = fmaxf(m, red[i][k]);
        red[0][k] = m;
    }
    __syncthreads();
    mx = red[0][k];
    __syncthreads();

    float s = 0.f;
    for (int l = g; l < L_SEQ; l += 8) s += __expf(Sb[(size_t)l * NCODES + k] - mx);
    red[g][k] = s;
    __syncthreads();
    if (g == 0) {
        float t = 0.f;
        for (int i = 0; i < 8; ++i) t += red[i][k];
        red[0][k] = 1.0f / t;
    }
    __syncthreads();
    const float inv = red[0][k];
    for (int l = g; l < L_SEQ; l += 8) {
        size_t o = (size_t)l * NCODES + k;
        Sb[o] = __expf(Sb[o] - mx) * inv;
    }
}

// ---- Kernel 3: Part[seg,b] = P(seg)^T @ X(seg)  (WMMA f32, double-buffered async LDS) ----
#define XBUF_FLTS (ROWS_C * LDSTR)    // one X buffer
#define PBUF_FLTS (ROWS_C * NCODES)   // one P buffer
#define PBASE_FLTS (2 * XBUF_FLTS)    // P buffers start after both X buffers

__global__ __launch_bounds__(256) void rc_context_kernel(const float* __restrict__ X,
                                                         const float* __restrict__ P,
                                                         float* __restrict__ Part) {
    extern __shared__ float smem[];

    const int b   = blockIdx.x;
    const int seg = blockIdx.y;         // 0..3, 512 rows each
    const int tid = threadIdx.x;
    const int wave = tid >> 5;
    const int lane = tid & 31;
    const int half = lane >> 4;
    const int lm   = lane & 15;
    const int d0   = wave * 32;         // this wave's 32 output columns

    const float* Xb = X + (size_t)b * L_SEQ * D_MODEL;
    const float* Pb = P + (size_t)b * L_SEQ * NCODES;

    auto stage = [&](int c) {
        const int l0  = seg * (ROWS_C * NCHUNK) + c * ROWS_C;
        const int buf = c & 1;
        float* xd = smem + buf * XBUF_FLTS;
        float* pd = smem + PBASE_FLTS + buf * PBUF_FLTS;
        for (int idx = tid; idx < ROWS_C * (D_MODEL / 4); idx += 256) {
            int r = idx >> 6, c4 = idx & 63;
            async_copy_b128(Xb + (size_t)(l0 + r) * D_MODEL + c4 * 4,
                            xd + r * LDSTR + c4 * 4);
        }
        for (int idx = tid; idx < ROWS_C * (NCODES / 4); idx += 256) {
            int r = idx >> 3, c4 = idx & 7;
            async_copy_b128(Pb + (size_t)(l0 + r) * NCODES + c4 * 4,
                            pd + r * NCODES + c4 * 4);
        }
    };

    v8f acc00 = {}, acc01 = {}, acc10 = {}, acc11 = {};   // [mtile][ntile]

    stage(0);
    wait_asynccnt0();
    __syncthreads();

    for (int c = 0; c < NCHUNK; ++c) {
        if (c + 1 < NCHUNK) stage(c + 1);   // overlap next chunk's copy with this compute

        const int buf = c & 1;
        const float* Xs = smem + buf * XBUF_FLTS;
        const float* Ps = smem + PBASE_FLTS + buf * PBUF_FLTS;
        for (int kk = 0; kk < ROWS_C; kk += 4) {
            const int r0 = kk + 2 * half;
            v2f a0, a1, b0, b1;
            // A = P^T: M=code, K=l.  a.x from row r0, a.y from row r0+1
            a0.x = Ps[r0 * NCODES + lm];            a0.y = Ps[(r0 + 1) * NCODES + lm];
            a1.x = Ps[r0 * NCODES + 16 + lm];       a1.y = Ps[(r0 + 1) * NCODES + 16 + lm];
            // B = X: K=l, N=d
            b0.x = Xs[r0 * LDSTR + d0 + lm];        b0.y = Xs[(r0 + 1) * LDSTR + d0 + lm];
            b1.x = Xs[r0 * LDSTR + d0 + 16 + lm];   b1.y = Xs[(r0 + 1) * LDSTR + d0 + 16 + lm];
            acc00 = wmma_f32(a0, b0, acc00);
            acc01 = wmma_f32(a0, b1, acc01);
            acc10 = wmma_f32(a1, b0, acc10);
            acc11 = wmma_f32(a1, b1, acc11);
        }

        wait_asynccnt0();   // chunk c+1 landed; all waves done with buf (c&1)
        __syncthreads();
    }

    float* Pg = Part + (((size_t)seg * BATCH + b) * NCODES) * D_MODEL;
    for (int r = 0; r < 8; ++r) {
        int m = r + half * 8;
        int n = d0 + lm;
        Pg[(size_t)m * D_MODEL + n]             = acc00[r];
        Pg[(size_t)m * D_MODEL + n + 16]        = acc01[r];
        Pg[(size_t)(m + 16) * D_MODEL + n]      = acc10[r];
        Pg[(size_t)(m + 16) * D_MODEL + n + 16] = acc11[r];
    }
}

// ---------------- Kernel 4: deterministic reduction of 4 partials ----------------
__global__ __launch_bounds__(256) void rc_reduce_kernel(const float* __restrict__ Part,
                                                        float* __restrict__ out) {
    const int total4 = BATCH * NCODES * D_MODEL / 4;
    int i = blockIdx.x * 256 + threadIdx.x;
    if (i < total4) {
        const float4* p = (const float4*)Part;
        float4 s0 = p[i];
        float4 s1 = p[i + total4];
        float4 s2 = p[i + 2 * total4];
        float4 s3 = p[i + 3 * total4];
        float4 r;
        r.x = (s0.x + s1.x) + (s2.x + s3.x);
        r.y = (s0.y + s1.y) + (s2.y + s3.y);
        r.z = (s0.z + s1.z) + (s2.z + s3.z);
        r.w = (s0.w + s1.w) + (s2.w + s3.w);
        ((float4*)out)[i] = r;
    }
}

extern "C" void kernel_launch(void* const* d_in, const int* in_sizes, int n_in,
                              void* d_out, int out_size, void* d_ws, size_t ws_size,
                              hipStream_t stream) {
    (void)in_sizes; (void)n_in; (void)out_size; (void)ws_size;
    const float* X = (const float*)d_in[0];   // (64, 2048, 256) f32
    const float* W = (const float*)d_in[1];   // (32, 256) f32
    float* out = (float*)d_out;               // (64, 32, 256) f32

    float* S    = (float*)d_ws;                               // 16 MB scores/probs
    float* Part = S + (size_t)BATCH * L_SEQ * NCODES;         // 8 MB partial contexts

    const size_t ldsA = (size_t)((ROWS_A + NCODES) * LDSTR) * sizeof(float);        // ~100 KB
    rc_scores_kernel<<<dim3(BATCH, L_SEQ / ROWS_A), 128, ldsA, stream>>>(X, W, S);

    rc_softmax_kernel<<<BATCH, 256, 0, stream>>>(S);

    const size_t ldsC = (size_t)(2 * ROWS_C * LDSTR + 2 * ROWS_C * NCODES) * sizeof(float); // ~146 KB
    rc_context_kernel<<<dim3(BATCH, 4), 256, ldsC, stream>>>(X, S, Part);

    const int total4 = BATCH * NCODES * D_MODEL / 4;
    rc_reduce_kernel<<<(total4 + 255) / 256, 256, 0, stream>>>(Part, out);
}